// EdgeAttnConv_83099027243482
// MI455X (gfx1250) — compile-verified
//
#include <hip/hip_runtime.h>

// ============================================================================
// EdgeAttnConv (GAT-like) for MI455X / gfx1250, wave32.
//
// Memory-bound op (~1.2 GB moved, 4.2 GFLOP): keep full fp32 precision and use
// V_WMMA_F32_16X16X4_F32 for both GEMMs. Avoid materializing e=[E,64]
// (410 MB) by (a) folding e·att_edge into w_eff = We@att_edge (16 floats) for
// the score pass, and (b) recomputing e via WMMA inside the message pass.
// z (25.6 MB) stays L2-resident (192 MB L2); prefetch hides gather latency.
// ============================================================================

typedef __attribute__((ext_vector_type(2))) float v2f;
typedef __attribute__((ext_vector_type(8))) float v8f;

#define NEGV (-1000000000.0f)

// Order-preserving float<->uint map for atomicMax on signed floats.
__device__ __forceinline__ unsigned enc_f32(float f) {
  unsigned u = __float_as_uint(f);
  return (u & 0x80000000u) ? ~u : (u | 0x80000000u);
}
__device__ __forceinline__ float dec_f32(unsigned v) {
  unsigned u = (v & 0x80000000u) ? (v & 0x7FFFFFFFu) : ~v;
  return __uint_as_float(u);
}

// ---------------------------------------------------------------------------
// K0: out = x (residual init); denom = 0; maxenc = enc(NEG)
// ---------------------------------------------------------------------------
__global__ void k_init(const float* __restrict__ x, float* __restrict__ out,
                       float* __restrict__ denom, unsigned* __restrict__ maxenc,
                       int N) {
  int total = N * 64;
  for (int i = blockIdx.x * blockDim.x + threadIdx.x; i < total;
       i += gridDim.x * blockDim.x) {
    out[i] = x[i];
    if (i < N) {
      denom[i] = 0.0f;
      maxenc[i] = enc_f32(NEGV);
    }
  }
}

// ---------------------------------------------------------------------------
// K1: z = x @ Wn   [N,64] = [N,64]@[64,64], fp32 WMMA 16x16x4, K-chain of 16.
// One wave per 16-row tile; 4 column tiles of 16, A frags reused across them.
// ---------------------------------------------------------------------------
__global__ void __launch_bounds__(256) k_node_gemm(
    const float* __restrict__ x, const float* __restrict__ Wn,
    float* __restrict__ z, int N) {
  int wave = (blockIdx.x * blockDim.x + threadIdx.x) >> 5;
  int lane = threadIdx.x & 31;
  int nRowTiles = (N + 15) >> 4;
  if (wave >= nRowTiles) return;  // wave-uniform exit (EXEC stays all-ones)

  int rBase = wave << 4;
  bool fullTile = (rBase + 16 <= N);  // wave-uniform
  int half = lane >> 4;               // which half-wave: selects K pair
  int l16 = lane & 15;

  // A fragment: row = rBase + l16; per step t, K = 4t + 2*half + {0,1}
  int arow = rBase + l16;
  if (arow >= N) arow = N - 1;  // clamp loads; tail stores guarded below
  v2f a[16];
#pragma unroll
  for (int t = 0; t < 16; ++t) {
    int k = (t << 2) + (half << 1);
    a[t].x = x[arow * 64 + k];
    a[t].y = x[arow * 64 + k + 1];
  }

#pragma unroll
  for (int ct = 0; ct < 4; ++ct) {
    int col = (ct << 4) + l16;
    v8f acc = {};
#pragma unroll
    for (int t = 0; t < 16; ++t) {
      int k = (t << 2) + (half << 1);
      v2f b;
      b.x = Wn[k * 64 + col];
      b.y = Wn[(k + 1) * 64 + col];
      acc = __builtin_amdgcn_wmma_f32_16x16x4_f32(
          false, a[t], false, b, (short)0, acc, false, false);
    }
    // C layout: element j of lane -> row j + 8*half, col = l16 (+col base)
    if (fullTile) {  // branchless fast path: one uniform branch, no exec churn
#pragma unroll
      for (int j = 0; j < 8; ++j) {
        int row = rBase + j + (half << 3);
        z[row * 64 + col] = acc[j];
      }
    } else {
#pragma unroll
      for (int j = 0; j < 8; ++j) {
        int row = rBase + j + (half << 3);
        if (row < N) z[row * 64 + col] = acc[j];
      }
    }
  }
}

// ---------------------------------------------------------------------------
// K2: w_eff[k] = sum_c We[k,c] * att_edge[c]   (16 values; tiny)
// ---------------------------------------------------------------------------
__global__ void k_weff(const float* __restrict__ We,
                       const float* __restrict__ att_edge,
                       float* __restrict__ weff) {
  int k = threadIdx.x;
  if (k < 16) {
    float s = 0.0f;
    for (int c = 0; c < 64; ++c) s += We[k * 64 + c] * att_edge[c];
    weff[k] = s;
  }
}

// ---------------------------------------------------------------------------
// K3: per-node attention scalars sA = z.att_src, sD = z.att_dst
// (z was just written -> L2-resident read)
// ---------------------------------------------------------------------------
__global__ void k_node_scores(const float* __restrict__ z,
                              const float* __restrict__ att_src,
                              const float* __restrict__ att_dst,
                              float* __restrict__ sA, float* __restrict__ sD,
                              int N) {
  int n = blockIdx.x * blockDim.x + threadIdx.x;
  if (n >= N) return;
  const float4* zr = (const float4*)(z + (size_t)n * 64);
  const float4* as4 = (const float4*)att_src;
  const float4* ad4 = (const float4*)att_dst;
  float a = 0.0f, d = 0.0f;
#pragma unroll
  for (int i = 0; i < 16; ++i) {
    float4 v = zr[i], s = as4[i], t = ad4[i];
    a += v.x * s.x + v.y * s.y + v.z * s.z + v.w * s.w;
    d += v.x * t.x + v.y * t.y + v.z * t.z + v.w * t.w;
  }
  sA[n] = a;
  sD[n] = d;
}

// ---------------------------------------------------------------------------
// K4: per-edge score = leaky_relu(sA[src]+sD[dst]+edge_attr.w_eff);
//     store score; segment max via atomicMax on encoded uint.
// ---------------------------------------------------------------------------
__global__ void k_score(const int* __restrict__ ei,
                        const float* __restrict__ ea,
                        const float* __restrict__ sA,
                        const float* __restrict__ sD,
                        const float* __restrict__ weff,
                        float* __restrict__ scoreBuf,
                        unsigned* __restrict__ maxenc, int E) {
  int e = blockIdx.x * blockDim.x + threadIdx.x;
  if (e >= E) return;
  int s = ei[e], d = ei[E + e];
  float acc = sA[s] + sD[d];
  const float4* er = (const float4*)(ea + (size_t)e * 16);
  const float4* w4 = (const float4*)weff;
#pragma unroll
  for (int i = 0; i < 4; ++i) {
    float4 v = er[i], w = w4[i];
    acc += v.x * w.x + v.y * w.y + v.z * w.z + v.w * w.w;
  }
  float sc = acc > 0.0f ? acc : 0.2f * acc;  // leaky_relu(0.2)
  scoreBuf[e] = sc;
  atomicMax(maxenc + d, enc_f32(sc));
}

// ---------------------------------------------------------------------------
// K5: numerator = exp(score - max[dst]); denom[dst] += numerator (atomic)
//     (scoreBuf is overwritten in place with the numerator)
// ---------------------------------------------------------------------------
__global__ void k_exp(const int* __restrict__ ei, float* __restrict__ num,
                      const unsigned* __restrict__ maxenc,
                      float* __restrict__ denom, int E) {
  int e = blockIdx.x * blockDim.x + threadIdx.x;
  if (e >= E) return;
  int d = ei[E + e];
  float m = dec_f32(maxenc[d]);
  float ex = __expf(num[e] - m);
  num[e] = ex;
  atomicAdd(denom + d, ex);
}

// ---------------------------------------------------------------------------
// K6: message + scatter.  Per 16-edge tile: recompute e = edge_attr@We with
// WMMA (K=16 -> 4 chained 16x16x4), then each lane's C element is one
// (edge,col) output:  out[dst[m], col] += (z[src[m], col] + e_val) * alpha[m].
// Block = 256 threads = 8 waves = 128 edges; alpha/src/dst staged in LDS.
// Gathered z rows are prefetched during the staging phase so their (L2)
// latency overlaps the WMMA recompute of e.
// ---------------------------------------------------------------------------
__global__ void __launch_bounds__(256) k_message(
    const int* __restrict__ ei, const float* __restrict__ ea,
    const float* __restrict__ We, const float* __restrict__ z,
    const float* __restrict__ num, const float* __restrict__ denom,
    float* __restrict__ out, int E) {
  __shared__ float s_alpha[128];
  __shared__ int s_src[128];
  __shared__ int s_dst[128];

  int tid = threadIdx.x;
  int blockBase = blockIdx.x * 128;

  if (tid < 128) {
    int e = blockBase + tid;
    if (e < E) {
      int d = ei[E + e];
      int sN = ei[e];
      s_alpha[tid] = num[e] / (denom[d] + 1e-8f);
      s_src[tid] = sN;
      s_dst[tid] = d;
      // Prefetch the z row this edge will gather (256 B = 2 lines).
      const float* zrow = z + (size_t)sN * 64;
      __builtin_prefetch(zrow, 0, 3);
      __builtin_prefetch(zrow + 32, 0, 3);
    } else {
      s_alpha[tid] = 0.0f;  // zero alpha -> contributes 0
      s_src[tid] = 0;
      s_dst[tid] = 0;
    }
  }
  __syncthreads();

  int wave = tid >> 5;
  int lane = tid & 31;
  int half = lane >> 4;
  int l16 = lane & 15;
  int tileBase = wave << 4;            // local edge index base
  int eBase = blockBase + tileBase;    // global edge index base
  if (eBase >= E) return;              // wave-uniform

  // A frags: 16 edges x K=16 of edge_attr
  int arow = eBase + l16;
  if (arow >= E) arow = E - 1;
  v2f a[4];
#pragma unroll
  for (int t = 0; t < 4; ++t) {
    int k = (t << 2) + (half << 1);
    a[t].x = ea[(size_t)arow * 16 + k];
    a[t].y = ea[(size_t)arow * 16 + k + 1];
  }

#pragma unroll
  for (int ct = 0; ct < 4; ++ct) {
    int col = (ct << 4) + l16;
    v8f acc = {};
#pragma unroll
    for (int t = 0; t < 4; ++t) {
      int k = (t << 2) + (half << 1);
      v2f b;
      b.x = We[k * 64 + col];
      b.y = We[(k + 1) * 64 + col];
      acc = __builtin_amdgcn_wmma_f32_16x16x4_f32(
          false, a[t], false, b, (short)0, acc, false, false);
    }
#pragma unroll
    for (int j = 0; j < 8; ++j) {
      int li = tileBase + j + (half << 3);  // local edge in block
      float al = s_alpha[li];
      int sN = s_src[li];
      int dN = s_dst[li];
      float val = (z[(size_t)sN * 64 + col] + acc[j]) * al;
      atomicAdd(out + (size_t)dN * 64 + col, val);
    }
  }
}

// ---------------------------------------------------------------------------
// Host launcher
// ---------------------------------------------------------------------------
extern "C" void kernel_launch(void* const* d_in, const int* in_sizes, int n_in,
                              void* d_out, int out_size, void* d_ws,
                              size_t ws_size, hipStream_t stream) {
  const float* x = (const float*)d_in[0];
  const int* ei = (const int*)d_in[1];
  const float* ea = (const float*)d_in[2];
  const float* Wn = (const float*)d_in[3];
  const float* We = (const float*)d_in[4];
  const float* att_src = (const float*)d_in[5];
  const float* att_dst = (const float*)d_in[6];
  const float* att_edge = (const float*)d_in[7];

  int N = in_sizes[0] / 64;  // x is [N, 64]
  int E = in_sizes[1] / 2;   // edge_index is [2, E]
  float* out = (float*)d_out;

  // Workspace carve-up (~34 MB): z[N*64], sA[N], sD[N], denom[N],
  // maxenc[N], num[E], weff[16]
  float* z = (float*)d_ws;
  float* sA = z + (size_t)N * 64;
  float* sD = sA + N;
  float* denom = sD + N;
  unsigned* maxenc = (unsigned*)(denom + N);
  float* num = (float*)(maxenc + N);
  float* weff = num + E;

  int total = N * 64;
  int ib = (total + 255) / 256;
  hipLaunchKernelGGL(k_init, dim3(ib), dim3(256), 0, stream, x, out, denom,
                     maxenc, N);

  int nRowTiles = (N + 15) / 16;
  int gb = (nRowTiles + 7) / 8;  // 8 waves per 256-thread block
  hipLaunchKernelGGL(k_node_gemm, dim3(gb), dim3(256), 0, stream, x, Wn, z, N);

  hipLaunchKernelGGL(k_weff, dim3(1), dim3(32), 0, stream, We, att_edge, weff);

  hipLaunchKernelGGL(k_node_scores, dim3((N + 255) / 256), dim3(256), 0, stream,
                     z, att_src, att_dst, sA, sD, N);

  hipLaunchKernelGGL(k_score, dim3((E + 255) / 256), dim3(256), 0, stream, ei,
                     ea, sA, sD, weff, num, maxenc, E);

  hipLaunchKernelGGL(k_exp, dim3((E + 255) / 256), dim3(256), 0, stream, ei,
                     num, maxenc, denom, E);

  hipLaunchKernelGGL(k_message, dim3((E + 127) / 128), dim3(256), 0, stream, ei,
                     ea, We, z, num, denom, out, E);
}